// RelationAttention_43344809951243
// MI455X (gfx1250) — compile-verified
//
#include <hip/hip_runtime.h>
#include <math.h>

typedef float v2f __attribute__((ext_vector_type(2)));
typedef float v8f __attribute__((ext_vector_type(8)));

#define B_ 4
#define L_ 1024
#define H_ 256
#define A_ 64
#define BL (B_ * L_)

// Fast single-instruction transcendentals where the gfx1250 backend exposes them.
#if __has_builtin(__builtin_amdgcn_tanhf)
__device__ __forceinline__ float fast_tanh(float x) { return __builtin_amdgcn_tanhf(x); }
#else
__device__ __forceinline__ float fast_tanh(float x) { return tanhf(x); }
#endif

// ---------------------------------------------------------------------------
// Kernel 1: TA = X * Wta^T, JA = X * Wja^T  via V_WMMA_F32_16X16X4_F32
// Block = 128 threads (4 waves). Block handles a 16-row tile of X (rows of
// the flattened [B*L, H] matrix). Wave w produces the 16-wide a-column slice
// [16w, 16w+16) for BOTH outputs, sharing the A-matrix fragment.
// ---------------------------------------------------------------------------
__global__ __launch_bounds__(128) void proj_wmma_kernel(
    const float* __restrict__ X,     // [B*L, H]
    const float* __restrict__ Wta,   // [A, H]
    const float* __restrict__ Wja,   // [A, H]
    float* __restrict__ ta,          // [B*L, A]
    float* __restrict__ ja)          // [B*L, A]
{
    const int wave = threadIdx.x >> 5;        // 0..3  -> n-tile
    const int lane = threadIdx.x & 31;
    const int half = lane >> 4;               // 0 or 1 (K-pair select)
    const int l15  = lane & 15;

    const int row0 = blockIdx.x * 16;         // 16 rows of [B*L, H]
    const int m    = l15;                     // A-fragment row (ISA layout)
    const int n    = wave * 16 + l15;         // B-fragment column = output a

    const float* xrow = X   + (size_t)(row0 + m) * H_;
    const float* wtr  = Wta + (size_t)n * H_;
    const float* wjr  = Wja + (size_t)n * H_;

    v8f acc_t = {};
    v8f acc_j = {};

    #pragma unroll 4
    for (int k0 = 0; k0 < H_; k0 += 4) {
        const int kk = k0 + half * 2;         // this lane's K pair
        // f32 A-matrix 16x4: lanes 0-15 hold K={k0,k0+1}, lanes 16-31 K={k0+2,k0+3}
        v2f a  = *(const v2f*)(xrow + kk);
        // f32 B-matrix 4x16: same striping with M->N (B[k][n] = W[n][k])
        v2f bt = *(const v2f*)(wtr  + kk);
        v2f bj = *(const v2f*)(wjr  + kk);
        acc_t = __builtin_amdgcn_wmma_f32_16x16x4_f32(
            false, a, false, bt, (short)0, acc_t, false, false);
        acc_j = __builtin_amdgcn_wmma_f32_16x16x4_f32(
            false, a, false, bj, (short)0, acc_j, false, false);
    }

    // C/D 16x16 f32 layout: VGPR r -> (M = r + 8*half, N = lane&15)
    const int ncol = wave * 16 + l15;
    #pragma unroll
    for (int r = 0; r < 8; ++r) {
        const int mrow = row0 + half * 8 + r;
        ta[(size_t)mrow * A_ + ncol] = acc_t[r];
        ja[(size_t)mrow * A_ + ncol] = acc_j[r];
    }
}

// ---------------------------------------------------------------------------
// Kernel 2: scores[b,i,j] = softmax_j( sum_a v[a]*tanh(ta[i,a]+ja[j,a]+bias[a]) )
// Block = 256 threads (8 waves); wave w owns row i0+w of batch b.
// ja is tiled through LDS (128 j-rows, padded stride 65 -> conflict-free:
// lane j hits bank (j+a)%64). Each lane keeps its 32 scores
// (j = tile*128 + jl*32 + lane) in registers; softmax is fully in-register
// with wave32 shuffle reductions.
// Hot loop per element: v_add_f32 + v_tanh_f32 (TRANS, co-executes) + v_fmac.
// ---------------------------------------------------------------------------
#define JT 128                 // j-rows per LDS tile
#define JP 65                  // padded row stride (bank-conflict-free)
#define RPB 8                  // i-rows per block (one per wave)

__global__ __launch_bounds__(256) void relattn_score_kernel(
    const float* __restrict__ ta,    // [B*L, A]
    const float* __restrict__ ja,    // [B*L, A]
    const float* __restrict__ bias,  // [A]
    const float* __restrict__ v,     // [A]
    float* __restrict__ out)         // [B, L, L]
{
    __shared__ float ja_s[JT * JP];          // 33280 B
    __shared__ float tb_s[RPB * A_];         // ta row + bias, per wave
    __shared__ float v_s[A_];

    const int t    = threadIdx.x;
    const int wave = t >> 5;
    const int lane = t & 31;

    const int b  = blockIdx.x / (L_ / RPB);
    const int i0 = (blockIdx.x % (L_ / RPB)) * RPB;

    // Stage (ta + bias) for this block's 8 rows, and v.
    {
        int idx = t;                          // 512 elements, 256 threads
        #pragma unroll
        for (int rep = 0; rep < 2; ++rep, idx += 256) {
            const int w = idx >> 6, a = idx & 63;
            tb_s[idx] = ta[(size_t)((b * L_) + i0 + w) * A_ + a] + bias[a];
        }
        if (t < A_) v_s[t] = v[t];
    }

    const float* ja_b = ja + (size_t)b * L_ * A_;
    float sc[32];                             // this lane's 1024/32 scores

    for (int tile = 0; tile < L_ / JT; ++tile) {
        __syncthreads();
        // Cooperative load of 128x64 ja tile: 8 float4 per thread.
        const float* src = ja_b + (size_t)tile * JT * A_;
        #pragma unroll
        for (int it = 0; it < 8; ++it) {
            const int idx = (it * 256 + t) * 4;     // flat float index < 8192
            const float4 val = *(const float4*)(src + idx);
            const int jl = idx >> 6, a = idx & 63;
            float* dst = &ja_s[jl * JP + a];
            dst[0] = val.x; dst[1] = val.y; dst[2] = val.z; dst[3] = val.w;
        }
        __syncthreads();

        float acc[4] = {0.f, 0.f, 0.f, 0.f};
        #pragma unroll 4
        for (int a = 0; a < A_; ++a) {
            const float tv = tb_s[wave * A_ + a];   // wave-uniform broadcast
            const float vv = v_s[a];                // broadcast
            #pragma unroll
            for (int jl = 0; jl < 4; ++jl) {
                const float jv = ja_s[(jl * 32 + lane) * JP + a];
                acc[jl] = fmaf(vv, fast_tanh(tv + jv), acc[jl]);
            }
        }
        #pragma unroll
        for (int jl = 0; jl < 4; ++jl) sc[tile * 4 + jl] = acc[jl];
    }

    // In-register softmax over the 1024 j's of this row (32 per lane).
    float mx = -INFINITY;
    #pragma unroll
    for (int k = 0; k < 32; ++k) mx = fmaxf(mx, sc[k]);
    #pragma unroll
    for (int off = 16; off > 0; off >>= 1)
        mx = fmaxf(mx, __shfl_xor(mx, off, 32));

    float sum = 0.f;
    #pragma unroll
    for (int k = 0; k < 32; ++k) { sc[k] = __expf(sc[k] - mx); sum += sc[k]; }
    #pragma unroll
    for (int off = 16; off > 0; off >>= 1)
        sum += __shfl_xor(sum, off, 32);
    const float inv = __builtin_amdgcn_rcpf(sum);   // v_rcp_f32

    float* orow = out + (size_t)(b * L_ + i0 + wave) * L_;
    #pragma unroll
    for (int tile = 0; tile < 8; ++tile)
        #pragma unroll
        for (int jl = 0; jl < 4; ++jl)
            orow[tile * JT + jl * 32 + lane] = sc[tile * 4 + jl] * inv;
}

// ---------------------------------------------------------------------------
extern "C" void kernel_launch(void* const* d_in, const int* in_sizes, int n_in,
                              void* d_out, int out_size, void* d_ws, size_t ws_size,
                              hipStream_t stream) {
    const float* X    = (const float*)d_in[0];   // [B,L,H]
    const float* Wta  = (const float*)d_in[1];   // [A,H]
    const float* Wja  = (const float*)d_in[2];   // [A,H]
    const float* bias = (const float*)d_in[3];   // [1,1,1,A]
    const float* v    = (const float*)d_in[4];   // [1,A]
    float* out = (float*)d_out;                  // [B,L,L]

    float* ta = (float*)d_ws;                    // [B*L, A]
    float* ja = ta + (size_t)BL * A_;            // [B*L, A]  (2 MB total)

    proj_wmma_kernel<<<BL / 16, 128, 0, stream>>>(X, Wta, Wja, ta, ja);
    relattn_score_kernel<<<BL / RPB, 256, 0, stream>>>(ta, ja, bias, v, out);
}